// Net_58969900974604
// MI455X (gfx1250) — compile-verified
//
#include <hip/hip_runtime.h>
#include <math.h>

typedef float v2f __attribute__((ext_vector_type(2)));
typedef float v8f __attribute__((ext_vector_type(8)));

#define FEAT_IN  512
#define FEAT_H   16
#define FEAT_OUT 40

// ---------------------------------------------------------------------------
// Degree / normalization
// ---------------------------------------------------------------------------
__global__ void k_fill_deg(float* __restrict__ deg, int n) {
    int i = blockIdx.x * blockDim.x + threadIdx.x;
    if (i < n) deg[i] = 1.0f;  // self loop contributes 1
}

__global__ void k_deg_count(const int* __restrict__ dst, float* __restrict__ deg, int nE) {
    int e = blockIdx.x * blockDim.x + threadIdx.x;
    if (e < nE) atomicAdd(&deg[dst[e]], 1.0f);
}

__global__ void k_dinv(const float* __restrict__ deg, float* __restrict__ dinv, int n) {
    int i = blockIdx.x * blockDim.x + threadIdx.x;
    if (i < n) dinv[i] = rsqrtf(deg[i]);  // deg >= 1 always
}

// ---------------------------------------------------------------------------
// GEMM1: xw1[n,16] = x[n,512] @ W1[512,16]  via V_WMMA_F32_16X16X4_F32
// One wave handles a 16-row tile; 8 waves (256 threads) per block.
// W1 staged in LDS, swizzled so each lane reads its (K,K+1) pair as one b64.
// ---------------------------------------------------------------------------
__global__ __launch_bounds__(256) void k_gemm1_wmma(const float* __restrict__ x,
                                                    const float* __restrict__ W1,
                                                    float* __restrict__ xw1, int n) {
    __shared__ float ldsW[FEAT_IN * FEAT_H];  // 32 KB, swizzled
    for (int i = threadIdx.x; i < FEAT_IN * FEAT_H; i += 256) {
        int k = i >> 4, nn = i & 15;
        // pair-major: ldsW[(k/2)*32 + nn*2 + (k&1)] = W1[k][nn]
        ldsW[((k >> 1) << 5) + (nn << 1) + (k & 1)] = W1[i];
    }
    __syncthreads();

    int lane = threadIdx.x & 31;
    int wave = threadIdx.x >> 5;
    int tile = blockIdx.x * 8 + wave;       // 16-row tile
    int row0 = tile * 16;
    if (row0 >= n) return;                  // wave-uniform exit (after barrier)

    int lm = lane & 15;                     // A: row-in-tile, B: column
    int hi = lane >> 4;                     // lane half selects K pair

    const float* xrow = x + (size_t)(row0 + lm) * FEAT_IN + 2 * hi;

    v8f acc = {};
    #pragma unroll 8
    for (int k = 0; k < FEAT_IN; k += 4) {
        // A frag: lane half 0 -> K=k,k+1 ; half 1 -> K=k+2,k+3 (row M=lm)
        v2f a = *(const v2f*)(xrow + k);
        // B frag: same K pair, column lm
        int kb = k + 2 * hi;
        v2f b = *(const v2f*)(ldsW + ((kb >> 1) << 5) + (lm << 1));
        acc = __builtin_amdgcn_wmma_f32_16x16x4_f32(
            false, a, false, b, (short)0, acc, false, false);
    }

    // C layout: VGPR j -> row (row0 + 8*hi + j), col lm
    float* out = xw1 + (size_t)(row0 + 8 * hi) * FEAT_H + lm;
    #pragma unroll
    for (int j = 0; j < 8; ++j) out[j * FEAT_H] = acc[j];
}

// ---------------------------------------------------------------------------
// Layer-1 aggregation
// ---------------------------------------------------------------------------
__global__ void k_self1(const float* __restrict__ xw1, const float* __restrict__ dinv,
                        float* __restrict__ h1, int n) {
    int i = blockIdx.x * blockDim.x + threadIdx.x;
    if (i >= n * FEAT_H) return;
    float di = dinv[i / FEAT_H];
    h1[i] = xw1[i] * di * di;               // self-loop contribution, also inits buffer
}

__global__ void k_scatter1(const int* __restrict__ src, const int* __restrict__ dst,
                           const float* __restrict__ dinv, const float* __restrict__ xw1,
                           float* __restrict__ h1, int nE) {
    int t = blockIdx.x * blockDim.x + threadIdx.x;
    int e = t >> 2;                          // 4 groups of 4 features
    if (e >= nE) return;
    int g = (t & 3) * 4;
    int s = src[e], d = dst[e];
    float norm = dinv[s] * dinv[d];
    float4 v = *(const float4*)(xw1 + (size_t)s * FEAT_H + g);
    float* o = h1 + (size_t)d * FEAT_H + g;
    atomicAdd(o + 0, v.x * norm);
    atomicAdd(o + 1, v.y * norm);
    atomicAdd(o + 2, v.z * norm);
    atomicAdd(o + 3, v.w * norm);
}

__global__ void k_bias_relu1(float* __restrict__ h1, const float* __restrict__ b1, int n) {
    int i = blockIdx.x * blockDim.x + threadIdx.x;
    if (i >= n * FEAT_H) return;
    float v = h1[i] + b1[i & 15];
    h1[i] = v > 0.0f ? v : 0.0f;
}

// ---------------------------------------------------------------------------
// GEMM2: xw2[n,40] = h1[n,16] @ W2[16,40]  (tiny; VALU with W2 in LDS)
// ---------------------------------------------------------------------------
__global__ __launch_bounds__(256) void k_gemm2(const float* __restrict__ h1,
                                               const float* __restrict__ W2,
                                               float* __restrict__ xw2, int n) {
    __shared__ float w[FEAT_H * FEAT_OUT];   // 640 floats
    for (int i = threadIdx.x; i < FEAT_H * FEAT_OUT; i += 256) w[i] = W2[i];
    __syncthreads();
    int node = blockIdx.x * blockDim.x + threadIdx.x;
    if (node >= n) return;
    const float4* hp = (const float4*)(h1 + (size_t)node * FEAT_H);
    float4 h0 = hp[0], h4 = hp[1], h8 = hp[2], hc = hp[3];
    float hv[16] = {h0.x, h0.y, h0.z, h0.w, h4.x, h4.y, h4.z, h4.w,
                    h8.x, h8.y, h8.z, h8.w, hc.x, hc.y, hc.z, hc.w};
    float* o = xw2 + (size_t)node * FEAT_OUT;
    #pragma unroll 4
    for (int f = 0; f < FEAT_OUT; ++f) {
        float acc = 0.0f;
        #pragma unroll
        for (int k = 0; k < FEAT_H; ++k) acc = fmaf(hv[k], w[k * FEAT_OUT + f], acc);
        o[f] = acc;
    }
}

// ---------------------------------------------------------------------------
// Layer-2 aggregation (accumulates directly in d_out)
// ---------------------------------------------------------------------------
__global__ void k_self2(const float* __restrict__ xw2, const float* __restrict__ dinv,
                        float* __restrict__ agg2, int n) {
    int i = blockIdx.x * blockDim.x + threadIdx.x;
    if (i >= n * FEAT_OUT) return;
    float di = dinv[i / FEAT_OUT];
    agg2[i] = xw2[i] * di * di;
}

__global__ void k_scatter2(const int* __restrict__ src, const int* __restrict__ dst,
                           const float* __restrict__ dinv, const float* __restrict__ xw2,
                           float* __restrict__ agg2, int nE) {
    int t = blockIdx.x * blockDim.x + threadIdx.x;
    int e = t / 10;                          // 10 groups of 4 features
    if (e >= nE) return;
    int g = (t - e * 10) * 4;
    int s = src[e], d = dst[e];
    float norm = dinv[s] * dinv[d];
    float4 v = *(const float4*)(xw2 + (size_t)s * FEAT_OUT + g);
    float* o = agg2 + (size_t)d * FEAT_OUT + g;
    atomicAdd(o + 0, v.x * norm);
    atomicAdd(o + 1, v.y * norm);
    atomicAdd(o + 2, v.z * norm);
    atomicAdd(o + 3, v.w * norm);
}

__global__ void k_bias_logsoftmax(float* __restrict__ out, const float* __restrict__ b2, int n) {
    int node = blockIdx.x * blockDim.x + threadIdx.x;
    if (node >= n) return;
    float* row = out + (size_t)node * FEAT_OUT;
    float v[FEAT_OUT];
    float m = -INFINITY;
    #pragma unroll 8
    for (int f = 0; f < FEAT_OUT; ++f) { v[f] = row[f] + b2[f]; m = fmaxf(m, v[f]); }
    float sum = 0.0f;
    #pragma unroll 8
    for (int f = 0; f < FEAT_OUT; ++f) sum += expf(v[f] - m);
    float l = logf(sum);
    #pragma unroll 8
    for (int f = 0; f < FEAT_OUT; ++f) row[f] = v[f] - m - l;
}

// ---------------------------------------------------------------------------
static inline int cdiv(long long a, int b) { return (int)((a + b - 1) / b); }

extern "C" void kernel_launch(void* const* d_in, const int* in_sizes, int n_in,
                              void* d_out, int out_size, void* d_ws, size_t ws_size,
                              hipStream_t stream) {
    const float* x  = (const float*)d_in[0];
    const int*   ei = (const int*)d_in[1];
    const float* W1 = (const float*)d_in[2];
    const float* b1 = (const float*)d_in[3];
    const float* W2 = (const float*)d_in[4];
    const float* b2 = (const float*)d_in[5];
    float* out = (float*)d_out;

    const int n  = in_sizes[0] / FEAT_IN;    // 100000
    const int nE = in_sizes[1] / 2;          // 3200000
    const int* src = ei;
    const int* dst = ei + nE;

    float* ws   = (float*)d_ws;
    float* deg  = ws;                                 // n
    float* dinv = deg + n;                            // n
    float* xw1  = dinv + n;                           // n*16
    float* h1   = xw1 + (size_t)n * FEAT_H;           // n*16
    float* xw2  = h1  + (size_t)n * FEAT_H;           // n*40
    // agg2 lives in d_out

    const int B = 256;

    // degrees & normalization
    k_fill_deg <<<cdiv(n, B), B, 0, stream>>>(deg, n);
    k_deg_count<<<cdiv(nE, B), B, 0, stream>>>(dst, deg, nE);
    k_dinv     <<<cdiv(n, B), B, 0, stream>>>(deg, dinv, n);

    // layer 1
    int tiles = cdiv(n, 16);
    k_gemm1_wmma<<<cdiv(tiles, 8), B, 0, stream>>>(x, W1, xw1, n);
    k_self1     <<<cdiv((long long)n * FEAT_H, B), B, 0, stream>>>(xw1, dinv, h1, n);
    k_scatter1  <<<cdiv((long long)nE * 4, B), B, 0, stream>>>(src, dst, dinv, xw1, h1, nE);
    k_bias_relu1<<<cdiv((long long)n * FEAT_H, B), B, 0, stream>>>(h1, b1, n);

    // layer 2
    k_gemm2   <<<cdiv(n, B), B, 0, stream>>>(h1, W2, xw2, n);
    k_self2   <<<cdiv((long long)n * FEAT_OUT, B), B, 0, stream>>>(xw2, dinv, out, n);
    k_scatter2<<<cdiv((long long)nE * 10, B), B, 0, stream>>>(src, dst, dinv, xw2, out, nE);
    k_bias_logsoftmax<<<cdiv(n, B), B, 0, stream>>>(out, b2, n);
}